// GATLayer_64355789963632
// MI455X (gfx1250) — compile-verified
//
#include <hip/hip_runtime.h>

#define N_NODES 4096
#define F_IN    512
#define F_OUT   256
#define N_HEADS 4
#define ALPHA   0.2f
#define NEGINF  -1e30f

typedef __attribute__((ext_vector_type(2))) float v2f;
typedef __attribute__((ext_vector_type(8))) float v8f;

// -------------------------------------------------------------------------
// Kernel 1: Wh[h,n,o] = sum_i x[n,i] * W[h,i,o]
// One wave computes a 16x64 tile of Wh[h] with V_WMMA_F32_16X16X4_F32,
// reusing the A fragment across 4 adjacent N-subtiles.
// -------------------------------------------------------------------------
__global__ __launch_bounds__(128) void gat_wh_gemm(const float* __restrict__ x,
                                                   const float* __restrict__ W,
                                                   float* __restrict__ Wh) {
  const int lane = threadIdx.x & 31;
  const int wave = threadIdx.x >> 5;
  int tile = blockIdx.x * 4 + wave;         // 4096 tiles total (1024/head)
  const int h = tile >> 10;
  tile &= 1023;
  const int m0 = (tile >> 2) * 16;          // row block
  const int n0 = (tile & 3) * 64;           // col block (64 wide)

  const int am  = lane & 15;                // A: m index / B: n index
  const int ak2 = (lane >> 4) * 2;          // k base for this half-wave

  v8f acc0 = {}, acc1 = {}, acc2 = {}, acc3 = {};
  const float* xrow = x + (size_t)(m0 + am) * F_IN + ak2;
  const float* Wp   = W + (size_t)h * F_IN * F_OUT;

  for (int k0 = 0; k0 < F_IN; k0 += 4) {
    v2f a = *(const v2f*)(xrow + k0);       // x[m0+am][k0+ak2 .. +1]
    const float* wb = Wp + (size_t)(k0 + ak2) * F_OUT + n0 + am;
    v2f b0, b1, b2, b3;
    b0.x = wb[0];   b0.y = wb[F_OUT];
    b1.x = wb[16];  b1.y = wb[F_OUT + 16];
    b2.x = wb[32];  b2.y = wb[F_OUT + 32];
    b3.x = wb[48];  b3.y = wb[F_OUT + 48];
    acc0 = __builtin_amdgcn_wmma_f32_16x16x4_f32(false, a, false, b0, (short)0, acc0, false, false);
    acc1 = __builtin_amdgcn_wmma_f32_16x16x4_f32(false, a, false, b1, (short)0, acc1, false, false);
    acc2 = __builtin_amdgcn_wmma_f32_16x16x4_f32(false, a, false, b2, (short)0, acc2, false, false);
    acc3 = __builtin_amdgcn_wmma_f32_16x16x4_f32(false, a, false, b3, (short)0, acc3, false, false);
  }

  // C/D layout: VGPR r holds (M = r + 8*(lane>>4), N = lane&15)
  float* out = Wh + (size_t)h * N_NODES * F_OUT;
  const int mb = (lane >> 4) * 8;
  const int nc = lane & 15;
#pragma unroll
  for (int r = 0; r < 8; ++r) {
    const size_t m = (size_t)(m0 + mb + r);
    out[m * F_OUT + n0 +      nc] = acc0[r];
    out[m * F_OUT + n0 + 16 + nc] = acc1[r];
    out[m * F_OUT + n0 + 32 + nc] = acc2[r];
    out[m * F_OUT + n0 + 48 + nc] = acc3[r];
  }
}

// -------------------------------------------------------------------------
// Kernel 2: s_src[h,n] = Wh[h,n,:] . a[h,0:256] ; s_dst with a[h,256:512]
// -------------------------------------------------------------------------
__global__ void gat_scores(const float* __restrict__ Wh,
                           const float* __restrict__ a,
                           float* __restrict__ s_src,
                           float* __restrict__ s_dst) {
  const int idx = blockIdx.x * blockDim.x + threadIdx.x;   // h*N + n
  if (idx >= N_HEADS * N_NODES) return;
  const int h = idx >> 12;
  const float* row = Wh + (size_t)idx * F_OUT;
  const float* as  = a + (size_t)h * 2 * F_OUT;
  const float* ad  = as + F_OUT;
  float ss = 0.f, sd = 0.f;
  for (int o = 0; o < F_OUT; ++o) {
    const float v = row[o];
    ss += v * as[o];
    sd += v * ad[o];
  }
  s_src[idx] = ss;
  s_dst[idx] = sd;
}

// -------------------------------------------------------------------------
// Kernel 3: fused masked softmax + attn @ Wh + head-concat + ELU.
// Block = 256 threads (8 waves), 16 rows of one head -> 16x256 output tile.
//  - Pass 1 streams each adj row ONCE, compressing it to a 1-bit mask in
//    LDS via wave32 ballot while computing the masked row max (leaky_relu
//    is monotone, so no Wh access needed).
//  - Pass 2 (denominator) and the P-tile build read the LDS bitmask, so
//    adj HBM traffic is 1x instead of 3x (64 MB/head @ 23.3 TB/s).
//  - The 16x64 probability tile is double-buffered in LDS: each iteration
//    builds tile t+1 (exp VALU) while WMMA consumes tile t, one barrier
//    per iteration; WMMA co-executes with VALU on CDNA5.
// -------------------------------------------------------------------------
#define BUILD_P(JB, BUF)                                                \
  {                                                                     \
    _Pragma("unroll")                                                   \
    for (int e4 = 0; e4 < 4; ++e4) {                                    \
      const int idx = e4 * 256 + t;                                     \
      const int pr  = idx >> 6;                                         \
      const int pj  = idx & 63;                                         \
      const int j   = (JB) + pj;                                        \
      const unsigned mk = smask[pr * 128 + (j >> 5)];                   \
      float p = 0.f;                                                    \
      if ((mk >> (j & 31)) & 1u) {                                      \
        float ev = ssrc[pr] + sj[j];                                    \
        ev = ev > 0.f ? ev : ALPHA * ev;                                \
        p = __expf(ev - smax[pr]) * sinv[pr];                           \
      }                                                                 \
      (BUF)[idx] = p;                                                   \
    }                                                                   \
  }

__global__ __launch_bounds__(256) void gat_attn(const float* __restrict__ adj,
                                                const float* __restrict__ Wh,
                                                const float* __restrict__ s_src,
                                                const float* __restrict__ s_dst,
                                                float* __restrict__ out) {
  __shared__ float    sP[2][16 * 64];     // double-buffered P tile (8 KB)
  __shared__ unsigned smask[16 * 128];    // 16 rows x 4096 bits       (8 KB)
  __shared__ float    red2[16 * 32];      // per-row wave reductions   (2 KB)
  __shared__ float    smax[16], sinv[16], ssrc[16];

  const int h  = blockIdx.x >> 8;
  const int i0 = (blockIdx.x & 255) * 16;
  const int t  = threadIdx.x;
  const int lane = t & 31;
  const int w    = t >> 5;                // wave 0..7

  if (t < 16) ssrc[t] = s_src[h * N_NODES + i0 + t];
  __syncthreads();

  const float* sj = s_dst + h * N_NODES;

  // ---- pass 1: stream adj once -> bitmask in LDS + masked row max ----
#pragma unroll
  for (int rr = 0; rr < 2; ++rr) {
    const int row = w * 2 + rr;           // each wave owns 2 rows
    const float* arow = adj + (size_t)(i0 + row) * N_NODES;
    const float sirow = ssrc[row];
    float m = NEGINF;
    for (int c = 0; c < 128; ++c) {
      const int j = c * 32 + lane;
      const float av = arow[j];
      const bool act = (av != 0.f);
      const unsigned mk = __builtin_amdgcn_ballot_w32(act);
      if (lane == 0) smask[row * 128 + c] = mk;
      if (act) {
        float e = sirow + sj[j];
        e = e > 0.f ? e : ALPHA * e;
        m = fmaxf(m, e);
      }
    }
    red2[row * 32 + lane] = m;
  }
  __syncthreads();
  if (t < 16) {
    float mm = red2[t * 32];
    for (int q = 1; q < 32; ++q) mm = fmaxf(mm, red2[t * 32 + q]);
    smax[t] = mm;
  }
  __syncthreads();

  // ---- pass 2: denominator from the LDS bitmask (no adj re-read) ----
#pragma unroll
  for (int rr = 0; rr < 2; ++rr) {
    const int row = w * 2 + rr;
    const float sirow = ssrc[row];
    const float mrow  = smax[row];
    float d = 0.f;
    for (int c = 0; c < 128; ++c) {
      const unsigned mk = smask[row * 128 + c];
      if ((mk >> lane) & 1u) {
        const int j = c * 32 + lane;
        float e = sirow + sj[j];
        e = e > 0.f ? e : ALPHA * e;
        d += __expf(e - mrow);
      }
    }
    red2[row * 32 + lane] = d;
  }
  __syncthreads();
  if (t < 16) {
    float dd = 0.f;
    for (int q = 0; q < 32; ++q) dd += red2[t * 32 + q];
    sinv[t] = dd > 0.f ? 1.f / dd : 0.f;  // fully-masked row -> zero attn
  }
  __syncthreads();

  // ---- stage 2: acc(16x256) += P(16xJ) @ Wh(Jx256), double-buffered ----
  const int nbase = w * 32;               // 8 waves x 32 output cols
  const int am    = lane & 15;
  const int ak2   = (lane >> 4) * 2;
  v8f acc0 = {}, acc1 = {};
  const float* Whh = Wh + (size_t)h * N_NODES * F_OUT;

  BUILD_P(0, sP[0]);
  __syncthreads();

  int cur = 0;
  for (int jb = 0; jb < N_NODES; jb += 64) {
    const int nxt = cur ^ 1;
    if (jb + 64 < N_NODES) {
      BUILD_P(jb + 64, sP[nxt]);          // build next tile (VALU/exp)
    }
    const float* sPc = sP[cur];
#pragma unroll
    for (int ks = 0; ks < 16; ++ks) {     // consume current tile (WMMA)
      v2f a = *(const v2f*)(&sPc[am * 64 + ks * 4 + ak2]);
      const float* wb = Whh + (size_t)(jb + ks * 4 + ak2) * F_OUT + nbase + am;
      v2f b0, b1;
      b0.x = wb[0];   b0.y = wb[F_OUT];
      b1.x = wb[16];  b1.y = wb[F_OUT + 16];
      acc0 = __builtin_amdgcn_wmma_f32_16x16x4_f32(false, a, false, b0, (short)0, acc0, false, false);
      acc1 = __builtin_amdgcn_wmma_f32_16x16x4_f32(false, a, false, b1, (short)0, acc1, false, false);
    }
    __syncthreads();
    cur = nxt;
  }

  // ---- epilogue: head concat + ELU ----
  const int mb = (lane >> 4) * 8;
  const int nc = lane & 15;
#pragma unroll
  for (int rr = 0; rr < 8; ++rr) {
    const size_t row = (size_t)(i0 + mb + rr);
    float v0 = acc0[rr];
    float v1 = acc1[rr];
    v0 = v0 > 0.f ? v0 : __expf(v0) - 1.f;
    v1 = v1 > 0.f ? v1 : __expf(v1) - 1.f;
    out[row * (N_HEADS * F_OUT) + h * F_OUT + nbase +      nc] = v0;
    out[row * (N_HEADS * F_OUT) + h * F_OUT + nbase + 16 + nc] = v1;
  }
}

// -------------------------------------------------------------------------
extern "C" void kernel_launch(void* const* d_in, const int* in_sizes, int n_in,
                              void* d_out, int out_size, void* d_ws, size_t ws_size,
                              hipStream_t stream) {
  const float* x   = (const float*)d_in[0];
  const float* adj = (const float*)d_in[1];
  const float* W   = (const float*)d_in[2];
  const float* a   = (const float*)d_in[3];
  float* out = (float*)d_out;

  float* Wh    = (float*)d_ws;                                   // 16 MB
  float* s_src = Wh + (size_t)N_HEADS * N_NODES * F_OUT;         // 64 KB
  float* s_dst = s_src + (size_t)N_HEADS * N_NODES;              // 64 KB

  gat_wh_gemm<<<1024, 128, 0, stream>>>(x, W, Wh);
  gat_scores<<<(N_HEADS * N_NODES + 255) / 256, 256, 0, stream>>>(Wh, a, s_src, s_dst);
  gat_attn<<<N_HEADS * (N_NODES / 16), 256, 0, stream>>>(adj, Wh, s_src, s_dst, out);
}